// ShapeConsistencyRegularizer_25074019074119
// MI455X (gfx1250) — compile-verified
//
#include <hip/hip_runtime.h>
#include <math.h>

typedef __attribute__((ext_vector_type(2))) float v2f;
typedef __attribute__((ext_vector_type(8))) float v8f;

#define KNN 20

// ---------------------------------------------------------------------------
// Fused distance-tile + top-K kernel.
// One wave (32 threads) per 16-row block. Loops over 128 column tiles of 16.
// Each tile: V_WMMA_F32_16X16X4_F32 computes (-2*A) x B  (K padded 3->4),
// giving d^2 = |p|^2 + |q|^2 - 2 p.q (clamped at 0). Selection runs on d^2
// (order-identical to d); sqrt is applied only to the KSEL winners at the
// end. Tile staged through LDS so each lane owns one row (2 lanes/row,
// 8 cols each). SELF_SKIP drops d(i,i)=0 (matches argsort[:,1:K+1]).
// ---------------------------------------------------------------------------
template <int KSEL, bool SELF_SKIP>
__global__ void __launch_bounds__(32)
knn_wmma_kernel(const float* __restrict__ rowPts,  // [B,N,3] A-side
                const float* __restrict__ colPts,  // [B,N,3] B-side
                int* __restrict__ outIdx,          // [B,N,KSEL] (KSEL==1: [B,N])
                float* __restrict__ outMean,       // [B,N] or unused
                int N)
{
  const int b       = blockIdx.y;
  const int rowBase = blockIdx.x * 16;
  const int lane    = threadIdx.x;
  const int lidx    = lane & 15;
  const int lhalf   = lane >> 4;

  const float* RP = rowPts + (size_t)b * N * 3;
  const float* CP = colPts + (size_t)b * N * 3;

  __shared__ float rowN[16];
  __shared__ float colN[16];
  __shared__ float tile[16][17];            // +1 pad: conflict-free row reads
  __shared__ float mD[16][2][KSEL];
  __shared__ int   mI[16][2][KSEL];

  // Row point for this lane (both half-waves load the same 16 rows).
  const int myRow = rowBase + lidx;
  const float px = RP[myRow * 3 + 0];
  const float py = RP[myRow * 3 + 1];
  const float pz = RP[myRow * 3 + 2];
  if (lhalf == 0) rowN[lidx] = px * px + py * py + pz * pz;

  // A operand (16x4 f32): lanes 0-15 -> K=0,1 ; lanes 16-31 -> K=2,3.
  // Pre-scaled by -2 so the WMMA result is already -2*p.q.
  v2f a;
  a.x = -2.0f * (lhalf ? pz : px);
  a.y = -2.0f * (lhalf ? 0.0f : py);

  float best[KSEL];
  int   bidx[KSEL];
#pragma unroll
  for (int k = 0; k < KSEL; ++k) { best[k] = 3.4e38f; bidx[k] = -1; }

  const int NT = N / 16;
  for (int tn = 0; tn < NT; ++tn) {
    const int colBase = tn * 16;
    const int cIdx    = colBase + lidx;
    const float qx = CP[cIdx * 3 + 0];
    const float qy = CP[cIdx * 3 + 1];
    const float qz = CP[cIdx * 3 + 2];
    if (tn + 1 < NT)
      __builtin_prefetch(&CP[(colBase + 16 + lidx) * 3], 0, 1);
    if (lhalf == 0) colN[lidx] = qx * qx + qy * qy + qz * qz;

    // B operand (4x16 f32): lanes 0-15 -> K=0,1 ; lanes 16-31 -> K=2,3.
    v2f bv;
    bv.x = lhalf ? qz : qx;
    bv.y = lhalf ? 0.0f : qy;
    __syncthreads();

    v8f c = {};
    c = __builtin_amdgcn_wmma_f32_16x16x4_f32(false, a, false, bv,
                                              (short)0, c, false, false);

    // C layout: c[v] -> row (v + 8*lhalf), col lidx. Store clamped d^2.
#pragma unroll
    for (int v = 0; v < 8; ++v) {
      const int m = v + 8 * lhalf;
      tile[m][lidx] = fmaxf(rowN[m] + colN[lidx] + c[v], 0.0f);
    }
    __syncthreads();

    // Selection view: lane owns row lidx, columns [lhalf*8, lhalf*8+8).
#pragma unroll
    for (int t = 0; t < 8; ++t) {
      const int co   = lhalf * 8 + t;
      const int gcol = colBase + co;
      const float d  = tile[lidx][co];          // squared distance
      if (SELF_SKIP && gcol == myRow) continue;
      if (d < best[KSEL - 1]) {
        best[KSEL - 1] = d;
        bidx[KSEL - 1] = gcol;
#pragma unroll
        for (int k = KSEL - 1; k > 0; --k) {
          if (best[k] < best[k - 1]) {
            const float td = best[k]; best[k] = best[k - 1]; best[k - 1] = td;
            const int   ti = bidx[k]; bidx[k] = bidx[k - 1]; bidx[k - 1] = ti;
          }
        }
      }
    }
    __syncthreads();
  }

  // Merge the two per-row lane lists via LDS.
#pragma unroll
  for (int k = 0; k < KSEL; ++k) {
    mD[lidx][lhalf][k] = best[k];
    mI[lidx][lhalf][k] = bidx[k];
  }
  __syncthreads();

  if (lhalf == 0) {
    const int gr = rowBase + lidx;
    if (KSEL == 1) {
      const int w = (mD[lidx][0][0] <= mD[lidx][1][0]) ? mI[lidx][0][0]
                                                       : mI[lidx][1][0];
      outIdx[(size_t)b * N + gr] = w;
    } else {
      int i0 = 0, i1 = 0;
      float sum = 0.0f;
      for (int k = 0; k < KSEL; ++k) {
        const float d0 = mD[lidx][0][i0];
        const float d1 = mD[lidx][1][i1];
        float d2;
        if (d0 <= d1) {
          outIdx[((size_t)b * N + gr) * KSEL + k] = mI[lidx][0][i0];
          d2 = d0; ++i0;
        } else {
          outIdx[((size_t)b * N + gr) * KSEL + k] = mI[lidx][1][i1];
          d2 = d1; ++i1;
        }
        sum += (d2 > 0.0f) ? sqrtf(d2) : 0.0f;   // sqrt only for winners
      }
      outMean[(size_t)b * N + gr] = sum / (float)KSEL;
    }
  }
}

// ---------------------------------------------------------------------------
// Per-point 3x3 covariance -> analytic eigen -> normal + curvature.
// cov = Sum(x x^T) - K * m m^T  (matches Sum((x-m)(x-m)^T) exactly).
// ---------------------------------------------------------------------------
__global__ void __launch_bounds__(256)
cov_normal_kernel(const float* __restrict__ pts, const int* __restrict__ knn,
                  float* __restrict__ nrm, float* __restrict__ curv, int N)
{
  const int b = blockIdx.y;
  const int i = blockIdx.x * blockDim.x + threadIdx.x;
  if (i >= N) return;

  const float* P  = pts + (size_t)b * N * 3;
  const int*   id = knn + ((size_t)b * N + i) * KNN;

  float sx = 0, sy = 0, sz = 0;
  float Sxx = 0, Sxy = 0, Sxz = 0, Syy = 0, Syz = 0, Szz = 0;
  for (int k = 0; k < KNN; ++k) {
    const int j = id[k];
    const float x = P[j * 3 + 0];
    const float y = P[j * 3 + 1];
    const float z = P[j * 3 + 2];
    sx += x; sy += y; sz += z;
    Sxx += x * x; Sxy += x * y; Sxz += x * z;
    Syy += y * y; Syz += y * z; Szz += z * z;
  }
  const float inv = 1.0f / (float)KNN;
  const float mx = sx * inv, my = sy * inv, mz = sz * inv;
  const float a00 = Sxx - (float)KNN * mx * mx;
  const float a01 = Sxy - (float)KNN * mx * my;
  const float a02 = Sxz - (float)KNN * mx * mz;
  const float a11 = Syy - (float)KNN * my * my;
  const float a12 = Syz - (float)KNN * my * mz;
  const float a22 = Szz - (float)KNN * mz * mz;

  const float tr = a00 + a11 + a22;
  const float q  = tr * (1.0f / 3.0f);
  const float p1 = a01 * a01 + a02 * a02 + a12 * a12;
  const float p2 = (a00 - q) * (a00 - q) + (a11 - q) * (a11 - q) +
                   (a22 - q) * (a22 - q) + 2.0f * p1;

  float lmin;
  float nx = 1.0f, ny = 0.0f, nz = 0.0f;
  if (p2 < 1e-24f) {
    lmin = q;                              // isotropic / degenerate
  } else {
    const float p   = sqrtf(p2 * (1.0f / 6.0f));
    const float ip  = 1.0f / p;
    const float b00 = (a00 - q) * ip, b11 = (a11 - q) * ip, b22 = (a22 - q) * ip;
    const float b01 = a01 * ip, b02 = a02 * ip, b12 = a12 * ip;
    float r = 0.5f * (b00 * (b11 * b22 - b12 * b12) -
                      b01 * (b01 * b22 - b12 * b02) +
                      b02 * (b01 * b12 - b11 * b02));
    r = fminf(1.0f, fmaxf(-1.0f, r));
    const float phi = acosf(r) * (1.0f / 3.0f);
    lmin = q + 2.0f * p * cosf(phi + 2.0943951023931953f);  // smallest eig

    // Eigenvector of lmin: largest cross product of rows of (A - lmin I).
    const float m00 = a00 - lmin, m11 = a11 - lmin, m22 = a22 - lmin;
    const float c0x = a01 * a12 - m11 * a02;     // r0 x r1
    const float c0y = a02 * a01 - m00 * a12;
    const float c0z = m00 * m11 - a01 * a01;
    const float c1x = a01 * m22 - a12 * a02;     // r0 x r2
    const float c1y = a02 * a02 - m00 * m22;
    const float c1z = m00 * a12 - a01 * a02;
    const float c2x = m11 * m22 - a12 * a12;     // r1 x r2
    const float c2y = a12 * a02 - a01 * m22;
    const float c2z = a01 * a12 - m11 * a02;
    const float n0 = c0x * c0x + c0y * c0y + c0z * c0z;
    const float n1 = c1x * c1x + c1y * c1y + c1z * c1z;
    const float n2 = c2x * c2x + c2y * c2y + c2z * c2z;
    float bx = c0x, by = c0y, bz = c0z, bn = n0;
    if (n1 > bn) { bx = c1x; by = c1y; bz = c1z; bn = n1; }
    if (n2 > bn) { bx = c2x; by = c2y; bz = c2z; bn = n2; }
    if (bn > 1e-30f) {
      const float il = 1.0f / fmaxf(sqrtf(bn), 1e-12f);
      nx = bx * il; ny = by * il; nz = bz * il;
    }
  }

  const size_t o = (size_t)b * N + i;
  nrm[o * 3 + 0] = nx;
  nrm[o * 3 + 1] = ny;
  nrm[o * 3 + 2] = nz;
  curv[o] = lmin / (tr + 1e-8f);
}

// ---------------------------------------------------------------------------
// Per-batch geometric prior (masked variances on y / z of pred points).
// ---------------------------------------------------------------------------
__global__ void __launch_bounds__(256)
prior_kernel(const float* __restrict__ pts, const int* __restrict__ cid,
             float* __restrict__ partials, int N)
{
  const int b   = blockIdx.x;
  const int tid = threadIdx.x;
  __shared__ float red[11][256];

  float acc[11];
#pragma unroll
  for (int k = 0; k < 11; ++k) acc[k] = 0.0f;

  const float* P = pts + (size_t)b * N * 3;
  for (int i = tid; i < N; i += 256) {
    const float y = P[i * 3 + 1];
    const float z = P[i * 3 + 2];
    if (y > -0.1f && y < 0.1f) { acc[0] += y; acc[1] += y * y; acc[2] += 1.0f; }
    if (z > 0.3f)              { acc[3] += z; acc[4] += z * z; acc[5] += 1.0f; }
    if (y > 0.2f)              { acc[6] += y; acc[7] += y * y; acc[8] += 1.0f; }
    if (y < -0.2f)             { acc[9] += 1.0f; }
    if (z > 0.3f && y > 0.0f)  { acc[10] += 1.0f; }
  }
#pragma unroll
  for (int k = 0; k < 11; ++k) red[k][tid] = acc[k];
  __syncthreads();
  for (int s = 128; s > 0; s >>= 1) {
    if (tid < s) {
#pragma unroll
      for (int k = 0; k < 11; ++k) red[k][tid] += red[k][tid + s];
    }
    __syncthreads();
  }

  if (tid == 0) {
    const float sS = red[0][0], sSS = red[1][0], sC = red[2][0];
    const float bS = red[3][0], bSS = red[4][0], bC = red[5][0];
    const float tS = red[6][0], tSS = red[7][0], tC = red[8][0];
    const float legs = red[9][0], sbc = red[10][0];

    const float sMean = sS / fmaxf(sC, 1.0f);
    const float sv = (sSS - 2.0f * sMean * sS + sMean * sMean * sC) / fmaxf(sC - 1.0f, 1.0f);
    const float bMean = bS / fmaxf(bC, 1.0f);
    const float bv = (bSS - 2.0f * bMean * bS + bMean * bMean * bC) / fmaxf(bC - 1.0f, 1.0f);
    const float tMean = tS / fmaxf(tC, 1.0f);
    const float tv = (tSS - 2.0f * tMean * tS + tMean * tMean * tC) / fmaxf(tC - 1.0f, 1.0f);

    const float chair = (sC > 0.0f ? sv * 10.0f : 0.0f) +
                        (bC > 0.0f ? fmaxf(0.05f - bv, 0.0f) * 5.0f : 0.0f);
    const float table = (tC > 0.0f ? tv * 10.0f : 0.0f) +
                        (legs < 10.0f ? 0.1f : 0.0f);
    const float sofa  = (sC > 0.0f ? sv * 5.0f : 0.0f) +
                        ((sbc < 10.0f || sbc / (float)N < 0.1f) ? 0.1f : 0.0f);

    const int c = cid[b];
    partials[b * 4 + 2] = (c == 2) ? chair : (c == 8) ? table : (c == 7) ? sofa : 0.0f;
  }
}

// ---------------------------------------------------------------------------
// Per-batch normal / structure / curvature loss means.
// ---------------------------------------------------------------------------
__global__ void __launch_bounds__(256)
loss_kernel(const float* __restrict__ pn, const float* __restrict__ gn,
            const float* __restrict__ pm, const float* __restrict__ gm,
            const float* __restrict__ pc, const float* __restrict__ gc,
            const int* __restrict__ mi, float* __restrict__ partials, int N)
{
  const int b   = blockIdx.x;
  const int tid = threadIdx.x;
  __shared__ float sN[256], sS[256], sC[256];

  float an = 0.0f, as = 0.0f, ac = 0.0f;
  for (int i = tid; i < N; i += 256) {
    const size_t ip = (size_t)b * N + i;
    const size_t jp = (size_t)b * N + mi[ip];
    const float dot = pn[ip * 3 + 0] * gn[jp * 3 + 0] +
                      pn[ip * 3 + 1] * gn[jp * 3 + 1] +
                      pn[ip * 3 + 2] * gn[jp * 3 + 2];
    an += fabsf(dot);
    const float cm = gm[jp];
    as += fabsf(pm[ip] - cm) / (cm + 1e-8f);
    ac += fabsf(pc[ip] - gc[jp]);
  }
  sN[tid] = an; sS[tid] = as; sC[tid] = ac;
  __syncthreads();
  for (int s = 128; s > 0; s >>= 1) {
    if (tid < s) { sN[tid] += sN[tid + s]; sS[tid] += sS[tid + s]; sC[tid] += sC[tid + s]; }
    __syncthreads();
  }
  if (tid == 0) {
    partials[b * 4 + 0] = 1.0f - sN[0] / (float)N;
    partials[b * 4 + 1] = sS[0] / (float)N;
    partials[b * 4 + 3] = sC[0] / (float)N;
  }
}

__global__ void final_kernel(const float* __restrict__ partials,
                             float* __restrict__ out, int B)
{
  if (threadIdx.x == 0 && blockIdx.x == 0) {
    float s = 0.0f;
    for (int b = 0; b < B; ++b)
      for (int q = 0; q < 4; ++q) s += partials[b * 4 + q];
    out[0] = s / (float)B;
  }
}

// ---------------------------------------------------------------------------
extern "C" void kernel_launch(void* const* d_in, const int* in_sizes, int n_in,
                              void* d_out, int out_size, void* d_ws, size_t ws_size,
                              hipStream_t stream) {
  const float* pred = (const float*)d_in[0];
  const float* gt   = (const float*)d_in[1];
  const int*   cid  = (const int*)d_in[2];

  const int B = in_sizes[2];
  const int N = in_sizes[0] / (B * 3);
  const size_t NB = (size_t)B * N;

  char* w = (char*)d_ws;
  int*   idxP = (int*)w;      w += NB * KNN * sizeof(int);
  int*   idxG = (int*)w;      w += NB * KNN * sizeof(int);
  int*   mi   = (int*)w;      w += NB * sizeof(int);
  float* pm   = (float*)w;    w += NB * sizeof(float);
  float* gm   = (float*)w;    w += NB * sizeof(float);
  float* pn   = (float*)w;    w += NB * 3 * sizeof(float);
  float* gn   = (float*)w;    w += NB * 3 * sizeof(float);
  float* pc   = (float*)w;    w += NB * sizeof(float);
  float* gc   = (float*)w;    w += NB * sizeof(float);
  float* partials = (float*)w;

  dim3 gK(N / 16, B);
  knn_wmma_kernel<KNN, true><<<gK, 32, 0, stream>>>(pred, pred, idxP, pm, N);
  knn_wmma_kernel<KNN, true><<<gK, 32, 0, stream>>>(gt,   gt,   idxG, gm, N);
  knn_wmma_kernel<1, false><<<gK, 32, 0, stream>>>(pred, gt,   mi, nullptr, N);

  dim3 gC((N + 255) / 256, B);
  cov_normal_kernel<<<gC, 256, 0, stream>>>(pred, idxP, pn, pc, N);
  cov_normal_kernel<<<gC, 256, 0, stream>>>(gt,   idxG, gn, gc, N);

  prior_kernel<<<B, 256, 0, stream>>>(pred, cid, partials, N);
  loss_kernel<<<B, 256, 0, stream>>>(pn, gn, pm, gm, pc, gc, mi, partials, N);
  final_kernel<<<1, 32, 0, stream>>>(partials, (float*)d_out, B);
}